// SlangGAT_13709535609054
// MI455X (gfx1250) — compile-verified
//
#include <hip/hip_runtime.h>
#include <math.h>

// ---------------------------------------------------------------------------
// 2-layer GAT for MI455X (gfx1250), wave32.
//  - GEMMs via V_WMMA_F32_16X16X4_F32 (full fp32 precision), W staged in LDS.
//  - A-fragment rows are clamped (not zero-filled): padded rows only affect
//    padded output rows which are never stored -> clean unpredicated b64 loads.
//  - Segment softmax via 3 edge sweeps: atomicMax (monotone uint encoding),
//    atomicAdd denom, atomicAdd messages (float4 gathers, 4 ch / thread).
//  - Whole working set is L2-resident (192 MB L2), so gather/atomic traffic
//    never leaves L2 after first touch.
// ---------------------------------------------------------------------------

typedef __attribute__((ext_vector_type(2))) float v2f;
typedef __attribute__((ext_vector_type(8))) float v8f;

#define NEG_SLOPE 0.2f

__device__ __forceinline__ unsigned fenc(float f) {
    unsigned u = __float_as_uint(f);
    return (u & 0x80000000u) ? ~u : (u | 0x80000000u);   // order-preserving
}
__device__ __forceinline__ float fdec(unsigned u) {
    return (u & 0x80000000u) ? __uint_as_float(u & 0x7fffffffu)
                             : __uint_as_float(~u);
}
__device__ __forceinline__ float lrelu(float x) { return x > 0.f ? x : NEG_SLOPE * x; }

__device__ __forceinline__ void edge_sd(const int* __restrict__ ei, int E, int e,
                                        int& s, int& d) {
    if (e < E) { s = ei[e]; d = ei[E + e]; }
    else       { s = d = e - E; }            // appended self-loops
}

// ------------------------- GEMM: [N,128] x [128, NCOL*16] -------------------
// One wave per 16-row tile; 8 waves / block; W in LDS.
template <int NCOL>
__global__ __launch_bounds__(256)
void gemm128_wmma(const float* __restrict__ X, const float* __restrict__ W,
                  float* __restrict__ H, int N) {
    constexpr int OW = NCOL * 16;
    __shared__ float sW[128 * OW];
    for (int i = threadIdx.x; i < 128 * OW; i += 256) sW[i] = W[i];
    __syncthreads();

    const int wid  = threadIdx.x >> 5;
    const int lane = threadIdx.x & 31;
    const int tile = blockIdx.x * 8 + wid;
    const int numTiles = (N + 15) >> 4;
    if (tile >= numTiles) return;

    const int half = lane >> 4;        // 0 or 1
    const int l16  = lane & 15;
    const int row  = tile * 16 + l16;
    // Clamp instead of zero-fill: D row M depends only on A row M, and
    // rows >= N are never stored, so garbage there is harmless.
    const int rowC = row < N ? row : N - 1;
    const float* xrow = X + (size_t)rowC * 128;

    v8f acc[NCOL] = {};
    for (int k0 = 0; k0 < 128; k0 += 4) {
        const int kb = k0 + (half << 1);           // even -> 8B-aligned
        const v2f a = *(const v2f*)(xrow + kb);    // single global_load_b64
#pragma unroll
        for (int t = 0; t < NCOL; t++) {
            const int col = t * 16 + l16;
            v2f b;
            b.x = sW[kb * OW + col];
            b.y = sW[(kb + 1) * OW + col];
            acc[t] = __builtin_amdgcn_wmma_f32_16x16x4_f32(
                false, a, false, b, (short)0, acc[t], false, false);
        }
    }

    if (tile * 16 + 16 <= N) {
        // full tile (always the case when N % 16 == 0): unguarded stores
#pragma unroll
        for (int t = 0; t < NCOL; t++) {
#pragma unroll
            for (int r = 0; r < 8; r++) {
                const int rr = tile * 16 + r + (half << 3);
                H[(size_t)rr * OW + t * 16 + l16] = acc[t][r];
            }
        }
    } else {
#pragma unroll
        for (int t = 0; t < NCOL; t++) {
#pragma unroll
            for (int r = 0; r < 8; r++) {
                const int rr = tile * 16 + r + (half << 3);
                if (rr < N) H[(size_t)rr * OW + t * 16 + l16] = acc[t][r];
            }
        }
    }
}

// -------------------- layer 1: per-node attention + init --------------------
__global__ void node_att1(const float* __restrict__ H1, const float* __restrict__ as_w,
                          const float* __restrict__ ad_w, float* __restrict__ as_,
                          float* __restrict__ ad_, unsigned* __restrict__ m,
                          float* __restrict__ den, float* __restrict__ out1, int N) {
    int idx = blockIdx.x * blockDim.x + threadIdx.x;
    if (idx >= N * 8) return;
    const int n = idx >> 3, h = idx & 7;
    const float* hp = H1 + (size_t)n * 128 + h * 16;
    float s = 0.f, d = 0.f;
#pragma unroll
    for (int i = 0; i < 16; i++) {
        const float v = hp[i];
        s += v * as_w[h * 16 + i];
        d += v * ad_w[h * 16 + i];
    }
    as_[idx] = s; ad_[idx] = d; m[idx] = 0u; den[idx] = 0.f;
    float4* o = (float4*)(out1 + (size_t)n * 128 + h * 16);
    const float4 z = {0.f, 0.f, 0.f, 0.f};
    o[0] = z; o[1] = z; o[2] = z; o[3] = z;
}

// ------------------------------ layer 1 edges -------------------------------
__global__ void edge_max1(const int* __restrict__ ei, int E, int EP,
                          const float* __restrict__ as_, const float* __restrict__ ad_,
                          unsigned* __restrict__ m) {
    int e = blockIdx.x * blockDim.x + threadIdx.x;
    if (e >= EP) return;
    int s, d; edge_sd(ei, E, e, s, d);
#pragma unroll
    for (int h = 0; h < 8; h++) {
        const float l = lrelu(as_[s * 8 + h] + ad_[d * 8 + h]);
        atomicMax(&m[d * 8 + h], fenc(l));
    }
}

__global__ void edge_sum1(const int* __restrict__ ei, int E, int EP,
                          const float* __restrict__ as_, const float* __restrict__ ad_,
                          const unsigned* __restrict__ m, float* __restrict__ den) {
    int e = blockIdx.x * blockDim.x + threadIdx.x;
    if (e >= EP) return;
    int s, d; edge_sd(ei, E, e, s, d);
#pragma unroll
    for (int h = 0; h < 8; h++) {
        const float l = lrelu(as_[s * 8 + h] + ad_[d * 8 + h]);
        atomicAdd(&den[d * 8 + h], expf(l - fdec(m[d * 8 + h])));
    }
}

// one thread per (edge, 4 consecutive channels) -> coalesced float4 gathers
__global__ void edge_msg1(const int* __restrict__ ei, int E, int EP,
                          const float* __restrict__ as_, const float* __restrict__ ad_,
                          const unsigned* __restrict__ m, const float* __restrict__ den,
                          const float* __restrict__ H1, float* __restrict__ out1) {
    int idx = blockIdx.x * blockDim.x + threadIdx.x;
    if (idx >= EP * 32) return;
    const int e = idx >> 5;
    const int c = (idx & 31) << 2;     // channel base, multiple of 4
    const int h = c >> 4;              // 16 channels per head
    int s, d; edge_sd(ei, E, e, s, d);
    const float l = lrelu(as_[s * 8 + h] + ad_[d * 8 + h]);
    const float alpha = expf(l - fdec(m[d * 8 + h])) / (den[d * 8 + h] + 1e-16f);
    const float4 hv = *(const float4*)(H1 + (size_t)s * 128 + c);
    float* o = out1 + (size_t)d * 128 + c;
    atomicAdd(o + 0, hv.x * alpha);
    atomicAdd(o + 1, hv.y * alpha);
    atomicAdd(o + 2, hv.z * alpha);
    atomicAdd(o + 3, hv.w * alpha);
}

// bias + ELU, writes h1b (overwrites h1 buffer)
__global__ void finalize1(const float* __restrict__ out1, const float* __restrict__ b1,
                          float* __restrict__ h1b, int N) {
    int idx = blockIdx.x * blockDim.x + threadIdx.x;
    if (idx >= N * 32) return;
    const int n = idx >> 5, c = (idx & 31) << 2;
    float4 v = *(const float4*)(out1 + (size_t)n * 128 + c);
    v.x += b1[c];     v.y += b1[c + 1]; v.z += b1[c + 2]; v.w += b1[c + 3];
    v.x = v.x > 0.f ? v.x : expm1f(v.x);
    v.y = v.y > 0.f ? v.y : expm1f(v.y);
    v.z = v.z > 0.f ? v.z : expm1f(v.z);
    v.w = v.w > 0.f ? v.w : expm1f(v.w);
    *(float4*)(h1b + (size_t)n * 128 + c) = v;
}

// -------------------- layer 2: per-node attention + init --------------------
__global__ void node_att2(const float* __restrict__ H2, const float* __restrict__ as_w,
                          const float* __restrict__ ad_w, float* __restrict__ as_,
                          float* __restrict__ ad_, unsigned* __restrict__ m,
                          float* __restrict__ den, float* __restrict__ out2, int N) {
    int n = blockIdx.x * blockDim.x + threadIdx.x;
    if (n >= N) return;
    float s = 0.f, d = 0.f;
    for (int i = 0; i < 64; i++) {
        const float v = H2[(size_t)n * 64 + i];
        s += v * as_w[i];
        d += v * ad_w[i];
        out2[(size_t)n * 64 + i] = 0.f;
    }
    as_[n] = s; ad_[n] = d; m[n] = 0u; den[n] = 0.f;
}

// ------------------------------ layer 2 edges -------------------------------
__global__ void edge_max2(const int* __restrict__ ei, int E, int EP,
                          const float* __restrict__ as_, const float* __restrict__ ad_,
                          unsigned* __restrict__ m) {
    int e = blockIdx.x * blockDim.x + threadIdx.x;
    if (e >= EP) return;
    int s, d; edge_sd(ei, E, e, s, d);
    atomicMax(&m[d], fenc(lrelu(as_[s] + ad_[d])));
}

__global__ void edge_sum2(const int* __restrict__ ei, int E, int EP,
                          const float* __restrict__ as_, const float* __restrict__ ad_,
                          const unsigned* __restrict__ m, float* __restrict__ den) {
    int e = blockIdx.x * blockDim.x + threadIdx.x;
    if (e >= EP) return;
    int s, d; edge_sd(ei, E, e, s, d);
    atomicAdd(&den[d], expf(lrelu(as_[s] + ad_[d]) - fdec(m[d])));
}

__global__ void edge_msg2(const int* __restrict__ ei, int E, int EP,
                          const float* __restrict__ as_, const float* __restrict__ ad_,
                          const unsigned* __restrict__ m, const float* __restrict__ den,
                          const float* __restrict__ H2, float* __restrict__ out2) {
    int idx = blockIdx.x * blockDim.x + threadIdx.x;
    if (idx >= EP * 16) return;
    const int e = idx >> 4;
    const int c = (idx & 15) << 2;
    int s, d; edge_sd(ei, E, e, s, d);
    const float l = lrelu(as_[s] + ad_[d]);
    const float alpha = expf(l - fdec(m[d])) / (den[d] + 1e-16f);
    const float4 hv = *(const float4*)(H2 + (size_t)s * 64 + c);
    float* o = out2 + (size_t)d * 64 + c;
    atomicAdd(o + 0, hv.x * alpha);
    atomicAdd(o + 1, hv.y * alpha);
    atomicAdd(o + 2, hv.z * alpha);
    atomicAdd(o + 3, hv.w * alpha);
}

// ------------------------- bias + log_softmax(64) ---------------------------
__global__ void logsoftmax64(const float* __restrict__ out2, const float* __restrict__ b2,
                             float* __restrict__ out, int N) {
    int n = blockIdx.x * blockDim.x + threadIdx.x;
    if (n >= N) return;
    const float* r = out2 + (size_t)n * 64;
    float mx = -INFINITY;
    for (int i = 0; i < 64; i++) { const float v = r[i] + b2[i]; mx = v > mx ? v : mx; }
    float ssum = 0.f;
    for (int i = 0; i < 64; i++) ssum += expf(r[i] + b2[i] - mx);
    const float lse = logf(ssum) + mx;
    for (int i = 0; i < 64; i++) out[(size_t)n * 64 + i] = r[i] + b2[i] - lse;
}

// ---------------------------------------------------------------------------
extern "C" void kernel_launch(void* const* d_in, const int* in_sizes, int n_in,
                              void* d_out, int out_size, void* d_ws, size_t ws_size,
                              hipStream_t stream) {
    const float* x      = (const float*)d_in[0];
    const int*   ei     = (const int*)  d_in[1];
    const float* W1     = (const float*)d_in[2];
    const float* att_s1 = (const float*)d_in[3];
    const float* att_d1 = (const float*)d_in[4];
    const float* b1     = (const float*)d_in[5];
    const float* W2     = (const float*)d_in[6];
    const float* att_s2 = (const float*)d_in[7];
    const float* att_d2 = (const float*)d_in[8];
    const float* b2     = (const float*)d_in[9];
    float* out = (float*)d_out;

    const int N  = in_sizes[0] / 128;
    const int E  = in_sizes[1] / 2;
    const int EP = E + N;                       // with self-loops

    // workspace layout (floats); buf0/buf1 are reused across phases
    float* ws = (float*)d_ws;
    float*    h1   = ws;                                  // [N,128]  buf0
    float*    out1 = h1 + (size_t)N * 128;                // [N,128]  buf1
    float*    as1  = out1 + (size_t)N * 128;              // [N,8]
    float*    ad1  = as1 + (size_t)N * 8;                 // [N,8]
    unsigned* m1   = (unsigned*)(ad1 + (size_t)N * 8);    // [N,8]
    float*    den1 = (float*)(m1 + (size_t)N * 8);        // [N,8]
    float*    as2  = den1 + (size_t)N * 8;                // [N]
    float*    ad2  = as2 + N;                             // [N]
    unsigned* m2   = (unsigned*)(ad2 + N);                // [N]
    float*    den2 = (float*)(m2 + N);                    // [N]
    float*    h1b  = h1;    // ELU output overwrites h1 (h1 consumed by msg1)
    float*    h2   = out1;  // layer-2 features overwrite out1
    float*    out2 = h1;    // layer-2 accumulator reuses buf0

    const int numTiles   = (N + 15) / 16;
    const int gemmBlocks = (numTiles + 7) / 8;

    // ---- layer 1 ----
    gemm128_wmma<8><<<gemmBlocks, 256, 0, stream>>>(x, W1, h1, N);
    node_att1<<<(N * 8 + 255) / 256, 256, 0, stream>>>(h1, att_s1, att_d1,
                                                       as1, ad1, m1, den1, out1, N);
    edge_max1<<<(EP + 255) / 256, 256, 0, stream>>>(ei, E, EP, as1, ad1, m1);
    edge_sum1<<<(EP + 255) / 256, 256, 0, stream>>>(ei, E, EP, as1, ad1, m1, den1);
    edge_msg1<<<(EP * 32 + 255) / 256, 256, 0, stream>>>(ei, E, EP, as1, ad1, m1,
                                                         den1, h1, out1);
    finalize1<<<(N * 32 + 255) / 256, 256, 0, stream>>>(out1, b1, h1b, N);

    // ---- layer 2 ----
    gemm128_wmma<4><<<gemmBlocks, 256, 0, stream>>>(h1b, W2, h2, N);
    node_att2<<<(N + 255) / 256, 256, 0, stream>>>(h2, att_s2, att_d2,
                                                   as2, ad2, m2, den2, out2, N);
    edge_max2<<<(EP + 255) / 256, 256, 0, stream>>>(ei, E, EP, as2, ad2, m2);
    edge_sum2<<<(EP + 255) / 256, 256, 0, stream>>>(ei, E, EP, as2, ad2, m2, den2);
    edge_msg2<<<(EP * 16 + 255) / 256, 256, 0, stream>>>(ei, E, EP, as2, ad2, m2,
                                                         den2, h2, out2);
    logsoftmax64<<<(N + 255) / 256, 256, 0, stream>>>(out2, b2, out, N);
}